// SSIMLoss_84748294685018
// MI455X (gfx1250) — compile-verified
//
#include <hip/hip_runtime.h>

typedef float v2f __attribute__((ext_vector_type(2)));
typedef float v8f __attribute__((ext_vector_type(8)));

// Separable factors of the gaussian: G = g^T g, g = [P, Q, P]
// P*P = 0.03797616, P*Q = 0.044863533, Q*Q = 0.053 (rank-1 to float precision)
#define GP 0.19487470f
#define GQ 0.23021729f

#define IMG   1024
#define CHS   (1024 * 1024)       // channel stride
#define TILE  64                  // output tile per block
#define HALO  8                   // max passes = 8
#define BW    80                  // LDS buffer rows/cols = TILE + 2*HALO
#define BST   81                  // LDS stride (odd -> conflict-free strided access)
#define NTHR  256

__device__ __forceinline__ int clampi(int v, int lo, int hi) {
    return v < lo ? lo : (v > hi ? hi : v);
}

// Band matrix entry: coefficient of input index r for output index m (both
// relative to the 16-tile origin; r already has the -2 K-window shift applied).
__device__ __forceinline__ float bandc(int m, int r) {
    int d = r - m;
    if (d == 0) return GQ;
    if (d == 1 || d == -1) return GP;
    return 0.0f;
}

// Load channel-sum of one image tile (+halo, zero padded at image edges) into
// bufA, then apply `passes` 3x3 gaussian convs with SAME zero padding.
// Each pass = vertical band-matmul (bufA->bufB) + horizontal band-matmul
// (bufB->bufA), both via V_WMMA_F32_16X16X4_F32 on 16x16 tiles (K=20 window
// in 5 chunks). Valid region erodes by 1/pass from the buffer edge; with
// HALO*2 = 16 margin and passes <= 8, the central 64x64 stays exact.
__device__ void conv_chain(const float* __restrict__ img,
                           float* bufA, float* bufB,
                           int gx0, int gy0, int passes) {
    const int tid = threadIdx.x;

    // ---- init: cross-channel sum with global zero padding ----
    for (int idx = tid; idx < BW * BW; idx += NTHR) {
        int r = idx / BW, c = idx % BW;
        int gr = gy0 + r, gc = gx0 + c;
        float v = 0.0f;
        if (gr >= 0 && gr < IMG && gc >= 0 && gc < IMG) {
            size_t o = (size_t)gr * IMG + gc;
            v = img[o] + img[o + CHS] + img[o + 2 * CHS];
        }
        bufA[r * BST + c] = v;
    }
    __syncthreads();

    const int lane  = tid & 31;          // wave32
    const int laneM = lane & 15;
    const int kb    = (lane < 16) ? 0 : 2;   // K sub-offset per lane half
    const int mOff  = (lane < 16) ? 0 : 8;   // D-matrix M offset per lane half
    const int wave  = tid >> 5;              // 8 waves

    for (int p = 0; p < passes; ++p) {
        // ---- vertical stage: out[i][j] = sum_r Band[i][r] * in[r][j] ----
        // A = band (constants), B = data (K-major f32 layout)
        for (int t = wave; t < 25; t += 8) {
            int row0 = (t / 5) * 16, col0 = (t % 5) * 16;
            v8f acc = {0.f, 0.f, 0.f, 0.f, 0.f, 0.f, 0.f, 0.f};
            #pragma unroll
            for (int c = 0; c < 5; ++c) {
                v2f wf, df;
                wf.x = bandc(laneM, 4 * c + kb - 2);
                wf.y = bandc(laneM, 4 * c + kb - 1);
                int rA  = row0 + 4 * c + kb - 2;
                int col = col0 + laneM;
                df.x = bufA[clampi(rA,     0, BW - 1) * BST + col];
                df.y = bufA[clampi(rA + 1, 0, BW - 1) * BST + col];
                acc = __builtin_amdgcn_wmma_f32_16x16x4_f32(
                        false, wf, false, df, (short)0, acc, false, false);
            }
            #pragma unroll
            for (int r = 0; r < 8; ++r)
                bufB[(row0 + mOff + r) * BST + col0 + laneM] = acc[r];
        }
        __syncthreads();

        // ---- horizontal stage: out[i][j] = sum_c in[i][c] * Band[c][j] ----
        // A = data (16x4 f32 layout), B = band (constants)
        for (int t = wave; t < 25; t += 8) {
            int row0 = (t / 5) * 16, col0 = (t % 5) * 16;
            v8f acc = {0.f, 0.f, 0.f, 0.f, 0.f, 0.f, 0.f, 0.f};
            #pragma unroll
            for (int c = 0; c < 5; ++c) {
                v2f wf, df;
                wf.x = bandc(laneM, 4 * c + kb - 2);
                wf.y = bandc(laneM, 4 * c + kb - 1);
                int cA  = col0 + 4 * c + kb - 2;
                int row = row0 + laneM;
                df.x = bufB[row * BST + clampi(cA,     0, BW - 1)];
                df.y = bufB[row * BST + clampi(cA + 1, 0, BW - 1)];
                acc = __builtin_amdgcn_wmma_f32_16x16x4_f32(
                        false, df, false, wf, (short)0, acc, false, false);
            }
            #pragma unroll
            for (int r = 0; r < 8; ++r) {
                int br = row0 + mOff + r, bc = col0 + laneM;
                int gr = gy0 + br, gc = gx0 + bc;
                bool inside = (gr >= 0) && (gr < IMG) && (gc >= 0) && (gc < IMG);
                bufA[br * BST + bc] = inside ? acc[r] : 0.0f;  // re-assert SAME zero pad
            }
        }
        __syncthreads();
    }
}

__global__ void __launch_bounds__(NTHR)
ssim_conv_kernel(const float* __restrict__ x, const float* __restrict__ y,
                 double* __restrict__ partial, int passes, float scale) {
    __shared__ float  bufA[BW * BST];
    __shared__ float  bufB[BW * BST];
    __shared__ double red[NTHR];

    const int tid = threadIdx.x;
    const int gx0 = blockIdx.x * TILE - HALO;
    const int gy0 = blockIdx.y * TILE - HALO;

    // u = 3^b * G^(b+1) * channelsum(x[b])
    conv_chain(x, bufA, bufB, gx0, gy0, passes);
    float u[16];
    #pragma unroll
    for (int i = 0; i < 16; ++i) {
        int idx = tid + i * NTHR;
        int r = idx >> 6, c = idx & 63;
        u[i] = bufA[(HALO + r) * BST + HALO + c] * scale;
    }
    __syncthreads();

    // v likewise for y[b]
    conv_chain(y, bufA, bufB, gx0, gy0, passes);

    double su = 0, su2 = 0, sv = 0, sv2 = 0, suv = 0;
    #pragma unroll
    for (int i = 0; i < 16; ++i) {
        int idx = tid + i * NTHR;
        int r = idx >> 6, c = idx & 63;
        double uu = (double)u[i];
        double vv = (double)(bufA[(HALO + r) * BST + HALO + c] * scale);
        su += uu; su2 += uu * uu; sv += vv; sv2 += vv * vv; suv += uu * vv;
    }
    __syncthreads();

    double vals[5] = {su, su2, sv, sv2, suv};
    int bid = blockIdx.y * gridDim.x + blockIdx.x;
    for (int k = 0; k < 5; ++k) {
        red[tid] = vals[k];
        __syncthreads();
        for (int off = NTHR / 2; off > 0; off >>= 1) {
            if (tid < off) red[tid] += red[tid + off];
            __syncthreads();
        }
        if (tid == 0) partial[bid * 5 + k] = red[0];
        __syncthreads();
    }
}

__global__ void ssim_reduce_kernel(const double* __restrict__ partial,
                                   double* __restrict__ term) {
    __shared__ double sh[NTHR];
    const int tid = threadIdx.x;
    double tot[5];
    for (int k = 0; k < 5; ++k) {
        sh[tid] = partial[tid * 5 + k];
        __syncthreads();
        for (int off = NTHR / 2; off > 0; off >>= 1) {
            if (tid < off) sh[tid] += sh[tid + off];
            __syncthreads();
        }
        if (tid == 0) tot[k] = sh[0];
        __syncthreads();
    }
    if (tid == 0) {
        const double N  = 1048576.0;          // pixels per channel
        const double C1 = 6.5025;             // (0.01*255)^2
        const double C2 = 58.5225;            // (0.03*255)^2
        double mx = tot[0] / N, my = tot[2] / N;
        double vx  = (tot[1] - tot[0] * tot[0] / N) / (N - 1.0);  // ddof=1 var
        double vy  = (tot[3] - tot[2] * tot[2] / N) / (N - 1.0);
        double cov = (tot[4] - tot[0] * tot[2] / N) / (N - 1.0);
        term[0] = ((2.0 * mx * my + C1) * (2.0 * cov + C2)) /
                  ((mx * mx + my * my + C1) * (vx * vx + vy * vy + C2));
    }
}

__global__ void ssim_final_kernel(const double* __restrict__ terms,
                                  float* __restrict__ out) {
    if (threadIdx.x == 0) {
        double s = 0.0;
        for (int b = 0; b < 8; ++b) s += terms[b];
        out[0] = (float)s;
    }
}

extern "C" void kernel_launch(void* const* d_in, const int* in_sizes, int n_in,
                              void* d_out, int out_size, void* d_ws, size_t ws_size,
                              hipStream_t stream) {
    (void)in_sizes; (void)n_in; (void)out_size; (void)ws_size;
    const float* x = (const float*)d_in[0];
    const float* y = (const float*)d_in[1];
    double* partial = (double*)d_ws;                                   // 256*5 doubles
    double* terms   = (double*)((char*)d_ws + 256 * 5 * sizeof(double)); // 8 doubles

    dim3 grid(IMG / TILE, IMG / TILE);   // 16x16 = 256 blocks
    dim3 block(NTHR);

    float scale = 1.0f;                  // 3^b cross-channel factor
    for (int b = 0; b < 8; ++b) {
        const float* xb = x + (size_t)b * 3 * CHS;
        const float* yb = y + (size_t)b * 3 * CHS;
        ssim_conv_kernel<<<grid, block, 0, stream>>>(xb, yb, partial, b + 1, scale);
        ssim_reduce_kernel<<<1, NTHR, 0, stream>>>(partial, terms + b);
        scale *= 3.0f;
    }
    ssim_final_kernel<<<1, 1, 0, stream>>>(terms, (float*)d_out);
}